// PrismLoss_88373247082702
// MI455X (gfx1250) — compile-verified
//
#include <hip/hip_runtime.h>

// Problem constants (from reference): B=128, U=4, NBS=64, S=408
#define PB   128
#define PU   4
#define PNBS 64
#define PS   408
#define EM_W 0.01f

// Derived
#define NRAD   (PB*PU*PNBS*PS)      // 13,369,344 elements in rad/tgt arrays
#define NATT   (PB*PU*PS)           // 208,896 elements in atten arrays
#define NRAD4  (NRAD/4)             // 3,342,336 float4 chunks
#define NATT4  (NATT/4)             // 52,224 float4 chunks
#define S4     (PS/4)               // 102 float4 chunks per s-row

#define NBLOCKS 1024
#define NTHREADS 256

typedef __attribute__((ext_vector_type(2))) float v2f;
typedef __attribute__((ext_vector_type(8))) float v8f;

// ---------------------------------------------------------------------------
// Exact f32 32-lane sum using the CDNA5 matrix pipe.
// V_WMMA_F32_16X16X4_F32: A is 16x4 f32 (2 VGPRs: lanes 0-15 hold K=0/1,
// lanes 16-31 hold K=2/3), B is 4x16. With A VGPR0 = v, VGPR1 = 0 and B = all
// ones (layout-invariant), D[i,j] = v[i] + v[i+16]. C-layout: VGPR r, lanes
// 0-15 hold D[r, j], lanes 16-31 hold D[8+r, j]. Summing the 8 D VGPRs gives
// lane<16: sum(v[0..7]) + sum(v[16..23]); lane>=16: sum(v[8..15]) +
// sum(v[24..31]).  total = t[lane0] + t[lane16].  All math exact in f32 adds.
// EXEC must be all ones at the call site (callers invoke it convergently).
// ---------------------------------------------------------------------------
__device__ __forceinline__ float wave_reduce_sum(float v) {
#if __has_builtin(__builtin_amdgcn_wmma_f32_16x16x4_f32)
  v2f a = {v, 0.0f};
  v2f b = {1.0f, 1.0f};
  v8f c = {};
  v8f d = __builtin_amdgcn_wmma_f32_16x16x4_f32(
      /*neg_a=*/false, a, /*neg_b=*/false, b,
      /*c_mod=*/(short)0, c, /*reuse_a=*/false, /*reuse_b=*/false);
  float t = ((d[0] + d[1]) + (d[2] + d[3])) + ((d[4] + d[5]) + (d[6] + d[7]));
  return __shfl(t, 0, 32) + __shfl(t, 16, 32);
#else
  for (int off = 16; off > 0; off >>= 1) v += __shfl_down(v, off, 32);
  return __shfl(v, 0, 32);
#endif
}

// Block reduction: result valid on thread 0.
__device__ __forceinline__ float block_reduce_sum(float v) {
  __shared__ float sm[NTHREADS / 32];
  float w = wave_reduce_sum(v);
  const int lane = threadIdx.x & 31;
  const int wid  = threadIdx.x >> 5;
  if (lane == 0) sm[wid] = w;
  __syncthreads();
  float r = 0.0f;
  if (threadIdx.x == 0) {
#pragma unroll
    for (int i = 0; i < NTHREADS / 32; ++i) r += sm[i];
  }
  return r;
}

// ---------------------------------------------------------------------------
// Kernel 1: streaming partial reduction. Each block writes one float to ws.
// ---------------------------------------------------------------------------
__global__ __launch_bounds__(NTHREADS)
void prism_partial_kernel(const float* __restrict__ atten_re,
                          const float* __restrict__ atten_im,
                          const float* __restrict__ rad_re,
                          const float* __restrict__ rad_im,
                          const float* __restrict__ tgt_re,
                          const float* __restrict__ tgt_im,
                          const float* __restrict__ weights,
                          float* __restrict__ partials) {
  const unsigned tid = blockIdx.x * NTHREADS + threadIdx.x;
  const unsigned nth = gridDim.x * NTHREADS;

  const float4* __restrict__ rr4 = (const float4*)rad_re;
  const float4* __restrict__ ri4 = (const float4*)rad_im;
  const float4* __restrict__ tr4 = (const float4*)tgt_re;
  const float4* __restrict__ ti4 = (const float4*)tgt_im;
  const float4* __restrict__ ar4 = (const float4*)atten_re;
  const float4* __restrict__ ai4 = (const float4*)atten_im;
  const float4* __restrict__ w4  = (const float4*)weights;

  float acc_r = 0.0f;   // sum(se * w)
  float acc_e = 0.0f;   // sum(relu(|rad|-10)^2)
  float acc_a = 0.0f;   // sum(relu(|atten|-1)^2)

  // ---- main stream over (B,U,NBS,S): 214 MB, HBM-bandwidth bound ----
  for (unsigned c = tid; c < (unsigned)NRAD4; c += nth) {
    const unsigned row = c / S4;           // (b*U+u)*NBS + n   (magic-mul div)
    const unsigned s4  = c - row * S4;     // float4 index inside s-row
    const unsigned au  = row >> 6;         // NBS == 64
    const unsigned aidx = au * S4 + s4;

    // Prefetch next grid-stride tile of the four streaming arrays.
    const unsigned cn = c + nth;
    if (cn < (unsigned)NRAD4) {
      __builtin_prefetch((const void*)(rr4 + cn), 0, 1);
      __builtin_prefetch((const void*)(ri4 + cn), 0, 1);
      __builtin_prefetch((const void*)(tr4 + cn), 0, 1);
      __builtin_prefetch((const void*)(ti4 + cn), 0, 1);
    }

    const float4 rr = rr4[c];
    const float4 ri = ri4[c];
    const float4 tr = tr4[c];
    const float4 ti = ti4[c];
    const float4 ar = ar4[aidx];   // L2-resident (1.7 MB, reused 64x)
    const float4 ai = ai4[aidx];
    const float4 w  = w4[s4];      // L0-resident (1.6 KB)

#define PRISM_ELEM(AR, AI, RR, RI, TR, TI, W)                     \
    {                                                             \
      const float pre = (AR) * (RR) - (AI) * (RI);                \
      const float pim = (AR) * (RI) + (AI) * (RR);                \
      const float dre = pre - (TR);                               \
      const float dim_ = pim - (TI);                              \
      acc_r += (dre * dre + dim_ * dim_) * (W);                   \
      const float m2 = (RR) * (RR) + (RI) * (RI);                 \
      const float t = fmaxf(__builtin_amdgcn_sqrtf(m2) - 10.0f, 0.0f); \
      acc_e += t * t;                                             \
    }

    PRISM_ELEM(ar.x, ai.x, rr.x, ri.x, tr.x, ti.x, w.x)
    PRISM_ELEM(ar.y, ai.y, rr.y, ri.y, tr.y, ti.y, w.y)
    PRISM_ELEM(ar.z, ai.z, rr.z, ri.z, tr.z, ti.z, w.z)
    PRISM_ELEM(ar.w, ai.w, rr.w, ri.w, tr.w, ti.w, w.w)
#undef PRISM_ELEM
  }

  // ---- small atten EM term over (B,U,S) ----
  for (unsigned c = tid; c < (unsigned)NATT4; c += nth) {
    const float4 ar = ar4[c];
    const float4 ai = ai4[c];
#define ATT_ELEM(AR, AI)                                          \
    {                                                             \
      const float m2 = (AR) * (AR) + (AI) * (AI);                 \
      const float t = fmaxf(__builtin_amdgcn_sqrtf(m2) - 1.0f, 0.0f); \
      acc_a += t * t;                                             \
    }
    ATT_ELEM(ar.x, ai.x)
    ATT_ELEM(ar.y, ai.y)
    ATT_ELEM(ar.z, ai.z)
    ATT_ELEM(ar.w, ai.w)
#undef ATT_ELEM
  }

  // Loss is linear in the three sums: fold normalizations per-thread so the
  // whole block reduces a single value.
  const float inv_nr = 1.0f / (float)NRAD;
  const float inv_na = 1.0f / (float)NATT;
  const float v = acc_r + EM_W * (acc_e * inv_nr + acc_a * inv_na);

  const float r = block_reduce_sum(v);   // convergent: EXEC all ones for WMMA
  if (threadIdx.x == 0) partials[blockIdx.x] = r;
}

// ---------------------------------------------------------------------------
// Kernel 2: reduce the per-block partials to the final scalar.
// ---------------------------------------------------------------------------
__global__ __launch_bounds__(NTHREADS)
void prism_final_kernel(const float* __restrict__ partials, int n,
                        float* __restrict__ out) {
  float v = 0.0f;
  for (int i = threadIdx.x; i < n; i += NTHREADS) v += partials[i];
  const float r = block_reduce_sum(v);
  if (threadIdx.x == 0) out[0] = r;
}

// ---------------------------------------------------------------------------
// Host launcher.
// setup_inputs order: atten_re, atten_im, rad_re, rad_im, tgt_re, tgt_im,
// weights, positions, ue_positions, view_directions, bs_antenna_ids
// (last four unused by the loss).
// ---------------------------------------------------------------------------
extern "C" void kernel_launch(void* const* d_in, const int* in_sizes, int n_in,
                              void* d_out, int out_size, void* d_ws,
                              size_t ws_size, hipStream_t stream) {
  (void)in_sizes; (void)n_in; (void)out_size; (void)ws_size;
  const float* atten_re = (const float*)d_in[0];
  const float* atten_im = (const float*)d_in[1];
  const float* rad_re   = (const float*)d_in[2];
  const float* rad_im   = (const float*)d_in[3];
  const float* tgt_re   = (const float*)d_in[4];
  const float* tgt_im   = (const float*)d_in[5];
  const float* weights  = (const float*)d_in[6];

  float* partials = (float*)d_ws;            // NBLOCKS floats (4 KB)
  float* out      = (float*)d_out;           // 1 float

  prism_partial_kernel<<<NBLOCKS, NTHREADS, 0, stream>>>(
      atten_re, atten_im, rad_re, rad_im, tgt_re, tgt_im, weights, partials);
  prism_final_kernel<<<1, NTHREADS, 0, stream>>>(partials, NBLOCKS, out);
}